// ProxyMLS_26379689132773
// MI455X (gfx1250) — compile-verified
//
#include <hip/hip_runtime.h>
#include <math.h>

typedef __attribute__((ext_vector_type(2))) float v2f;
typedef __attribute__((ext_vector_type(8))) float v8f;

#define N_EMB 4096
#define C_CLS 8192
#define D_DIM 64
#define K_CAT 128            // concat K = 2*D
#define MARGIN_F 0.1f
#define ALPHA_F 32.0f

// Block tiling for the WMMA GEMM: 8 waves = 2 (M) x 4 (N)
// wave tile = 32(M) x 64(N) = 2x4 accumulators of 16x16
#define BLK_M 64
#define BLK_N 256

// order-preserving float<->uint map so unsigned atomicMin == float min
__device__ __forceinline__ unsigned enc_f32(float f) {
    unsigned u = __float_as_uint(f);
    return (u & 0x80000000u) ? ~u : (u | 0x80000000u);
}
__device__ __forceinline__ float dec_f32(unsigned u) {
    return (u & 0x80000000u) ? __uint_as_float(u ^ 0x80000000u)
                             : __uint_as_float(~u);
}

__global__ void k_init(float* gpos, float* gneg, int* counts, unsigned* minbits) {
    int j = blockIdx.x * blockDim.x + threadIdx.x;
    if (j < C_CLS) { gpos[j] = 0.0f; gneg[j] = 0.0f; counts[j] = 0; }
    if (j == 0) *minbits = enc_f32(3.0e38f);
}

// Per class j: inv_var, mu*inv_var into B' (C x 128), const_j = sum_d(mu^2*iv + logvar)
__global__ void k_prep_proxies(const float* __restrict__ mu,
                               const float* __restrict__ lv,
                               float* __restrict__ Bcat,
                               float* __restrict__ constv) {
    __shared__ float sred[D_DIM];
    int j = blockIdx.x, d = threadIdx.x;
    float l = lv[(size_t)j * D_DIM + d];
    float m = mu[(size_t)j * D_DIM + d];
    float iv = expf(-l);
    Bcat[(size_t)j * K_CAT + d]          = iv;
    Bcat[(size_t)j * K_CAT + D_DIM + d]  = m * iv;
    sred[d] = m * m * iv + l;
    __syncthreads();
    for (int s = 32; s > 0; s >>= 1) {
        if (d < s) sred[d] += sred[d + s];
        __syncthreads();
    }
    if (d == 0) constv[j] = sred[0];
}

// Per embedding i: A' = [-0.5*e^2, e]; also per-class sample counts
__global__ void k_prep_emb(const float* __restrict__ e,
                           const int* __restrict__ labels,
                           float* __restrict__ Acat,
                           int* __restrict__ counts) {
    int i = blockIdx.x, d = threadIdx.x;
    float v = e[(size_t)i * D_DIM + d];
    Acat[(size_t)i * K_CAT + d]         = -0.5f * v * v;
    Acat[(size_t)i * K_CAT + D_DIM + d] = v;
    if (d == 0) atomicAdd(&counts[labels[i]], 1);
}

// Shared fragment-pointer setup + GEMM body: computes 2x4 accumulators/wave.
// A fragment (16x4 f32): lanes 0-15 hold K=kk,kk+1 ; lanes 16-31 hold K=kk+2,kk+3.
// B fragment (4x16 f32): mirrored (lane = N, halves select K pair).
#define GEMM_TILE_BODY                                                                 \
    int lane = threadIdx.x & 31;                                                       \
    int wave = threadIdx.x >> 5;                                                       \
    int ll = lane & 15;                                                                \
    int lh = lane >> 4;                                                                \
    int m0 = blockIdx.y * BLK_M + (wave & 1) * 32;                                     \
    int c0 = blockIdx.x * BLK_N + (wave >> 1) * 64;                                    \
    const v2f* ap0 = (const v2f*)(Acat + (size_t)(m0 +  0 + ll) * K_CAT + 2 * lh);     \
    const v2f* ap1 = (const v2f*)(Acat + (size_t)(m0 + 16 + ll) * K_CAT + 2 * lh);     \
    const v2f* bp0 = (const v2f*)(Bcat + (size_t)(c0 +  0 + ll) * K_CAT + 2 * lh);     \
    const v2f* bp1 = (const v2f*)(Bcat + (size_t)(c0 + 16 + ll) * K_CAT + 2 * lh);     \
    const v2f* bp2 = (const v2f*)(Bcat + (size_t)(c0 + 32 + ll) * K_CAT + 2 * lh);     \
    const v2f* bp3 = (const v2f*)(Bcat + (size_t)(c0 + 48 + ll) * K_CAT + 2 * lh);     \
    v8f acc[2][4] = {};                                                                \
    _Pragma("unroll")                                                                  \
    for (int kk = 0; kk < K_CAT; kk += 4) {                                            \
        v2f a0 = ap0[kk >> 1];                                                         \
        v2f a1 = ap1[kk >> 1];                                                         \
        v2f b0 = bp0[kk >> 1];                                                         \
        v2f b1 = bp1[kk >> 1];                                                         \
        v2f b2 = bp2[kk >> 1];                                                         \
        v2f b3 = bp3[kk >> 1];                                                         \
        acc[0][0] = __builtin_amdgcn_wmma_f32_16x16x4_f32(false, a0, false, b0, (short)0, acc[0][0], false, false); \
        acc[0][1] = __builtin_amdgcn_wmma_f32_16x16x4_f32(false, a0, false, b1, (short)0, acc[0][1], false, false); \
        acc[0][2] = __builtin_amdgcn_wmma_f32_16x16x4_f32(false, a0, false, b2, (short)0, acc[0][2], false, false); \
        acc[0][3] = __builtin_amdgcn_wmma_f32_16x16x4_f32(false, a0, false, b3, (short)0, acc[0][3], false, false); \
        acc[1][0] = __builtin_amdgcn_wmma_f32_16x16x4_f32(false, a1, false, b0, (short)0, acc[1][0], false, false); \
        acc[1][1] = __builtin_amdgcn_wmma_f32_16x16x4_f32(false, a1, false, b1, (short)0, acc[1][1], false, false); \
        acc[1][2] = __builtin_amdgcn_wmma_f32_16x16x4_f32(false, a1, false, b2, (short)0, acc[1][2], false, false); \
        acc[1][3] = __builtin_amdgcn_wmma_f32_16x16x4_f32(false, a1, false, b3, (short)0, acc[1][3], false, false); \
    }

// ---- Pass 1: fp32 WMMA GEMM tiles, global min(score) -------------------------
__global__ __launch_bounds__(256) void k_pass1_min(const float* __restrict__ Acat,
                                                   const float* __restrict__ Bcat,
                                                   const float* __restrict__ constv,
                                                   unsigned* __restrict__ minbits) {
    GEMM_TILE_BODY

    float lm = 3.0e38f;
    #pragma unroll
    for (int s = 0; s < 4; s++) {
        float cst = constv[c0 + s * 16 + ll];
        #pragma unroll
        for (int u = 0; u < 2; u++) {
            #pragma unroll
            for (int v = 0; v < 8; v++) {
                float sc = acc[u][s][v] - 0.5f * cst;
                lm = fminf(lm, sc);
            }
        }
    }
    #pragma unroll
    for (int off = 16; off > 0; off >>= 1)
        lm = fminf(lm, __shfl_xor(lm, off, 32));
    if (lane == 0) atomicMin(minbits, enc_f32(lm));
}

__global__ void k_scalar_m(const unsigned* minbits, float* mval) {
    float smin = dec_f32(*minbits);
    *mval = -ALPHA_F * (smin - MARGIN_F);   // m = max(pos_e)
}

// ---- Pass 2: recompute tiles, exp, per-class pos/neg sums --------------------
__global__ __launch_bounds__(256) void k_pass2(const float* __restrict__ Acat,
                                               const float* __restrict__ Bcat,
                                               const float* __restrict__ constv,
                                               const int* __restrict__ labels,
                                               const float* __restrict__ mval,
                                               float* __restrict__ gpos,
                                               float* __restrict__ gneg) {
    __shared__ int   slab[BLK_M];
    __shared__ float sneg[BLK_N];
    {
        int t = threadIdx.x;
        if (t < BLK_M) slab[t] = labels[blockIdx.y * BLK_M + t];
        sneg[t] = 0.0f;
        __syncthreads();
    }

    GEMM_TILE_BODY

    float mm = *mval;
    #pragma unroll
    for (int s = 0; s < 4; s++) {
        int col = c0 + s * 16 + ll;
        float cst = constv[col];
        float negpart = 0.0f;
        #pragma unroll
        for (int u = 0; u < 2; u++) {
            int rlbase = (wave & 1) * 32 + u * 16 + (lh << 3);  // row-local base
            #pragma unroll
            for (int v = 0; v < 8; v++) {
                float sc = acc[u][s][v] - 0.5f * cst;            // score(i,j)
                int rl = rlbase + v;                             // D-layout: M = v + 8*half
                if (slab[rl] == col) {
                    float p = expf(-ALPHA_F * (sc - MARGIN_F) - mm);
                    atomicAdd(&gpos[col], p);                    // rare (~1 per block)
                } else {
                    negpart += expf(ALPHA_F * (sc + MARGIN_F) - mm);
                }
            }
        }
        negpart += __shfl_xor(negpart, 16, 32);                  // fold lane halves
        if (lh == 0)
            atomicAdd(&sneg[(wave >> 1) * 64 + s * 16 + ll], negpart);
    }
    __syncthreads();
    {
        int t = threadIdx.x;
        atomicAdd(&gneg[(size_t)blockIdx.x * BLK_N + t], sneg[t]);
    }
}

// ---- Final scalar reduction --------------------------------------------------
__global__ __launch_bounds__(256) void k_final(const float* __restrict__ gpos,
                                               const float* __restrict__ gneg,
                                               const int* __restrict__ counts,
                                               const float* __restrict__ mval,
                                               float* __restrict__ out) {
    __shared__ float s0[256], s1[256], s2[256];
    int t = threadIdx.x;
    float slp = 0.0f, sln = 0.0f, nv = 0.0f;
    for (int j = t; j < C_CLS; j += 256) {
        slp += log1pf(gpos[j]);
        sln += log1pf(gneg[j]);
        nv  += (counts[j] != 0) ? 1.0f : 0.0f;
    }
    s0[t] = slp; s1[t] = sln; s2[t] = nv;
    __syncthreads();
    for (int s = 128; s > 0; s >>= 1) {
        if (t < s) { s0[t] += s0[t + s]; s1[t] += s1[t + s]; s2[t] += s2[t + s]; }
        __syncthreads();
    }
    if (t == 0) {
        float m = *mval;
        float pos_term = (s0[0] + (float)C_CLS * m) / s2[0];
        float neg_term = s1[0] / (float)C_CLS + m;
        out[0] = pos_term + neg_term;
    }
}

extern "C" void kernel_launch(void* const* d_in, const int* in_sizes, int n_in,
                              void* d_out, int out_size, void* d_ws, size_t ws_size,
                              hipStream_t stream) {
    const float* emb    = (const float*)d_in[0];
    const int*   labels = (const int*)d_in[1];
    const float* mu     = (const float*)d_in[2];
    const float* lv     = (const float*)d_in[3];

    float* ws     = (float*)d_ws;
    float* Bcat   = ws;                                     // C*128
    float* Acat   = Bcat + (size_t)C_CLS * K_CAT;           // N*128
    float* constv = Acat + (size_t)N_EMB * K_CAT;           // C
    float* gpos   = constv + C_CLS;                         // C
    float* gneg   = gpos + C_CLS;                           // C
    int*   counts = (int*)(gneg + C_CLS);                   // C
    unsigned* minbits = (unsigned*)(counts + C_CLS);        // 1
    float* mval   = (float*)(minbits + 1);                  // 1

    k_init<<<C_CLS / 256, 256, 0, stream>>>(gpos, gneg, counts, minbits);
    k_prep_proxies<<<C_CLS, 64, 0, stream>>>(mu, lv, Bcat, constv);
    k_prep_emb<<<N_EMB, 64, 0, stream>>>(emb, labels, Acat, counts);

    dim3 grid(C_CLS / BLK_N, N_EMB / BLK_M);
    k_pass1_min<<<grid, 256, 0, stream>>>(Acat, Bcat, constv, minbits);
    k_scalar_m<<<1, 1, 0, stream>>>(minbits, mval);
    k_pass2<<<grid, 256, 0, stream>>>(Acat, Bcat, constv, labels, mval, gpos, gneg);
    k_final<<<1, 256, 0, stream>>>(gpos, gneg, counts, mval, (float*)d_out);
}